// GraphEncoder_23287312678935
// MI455X (gfx1250) — compile-verified
//
#include <hip/hip_runtime.h>
#include <hip/hip_bf16.h>

typedef __attribute__((ext_vector_type(2))) float v2f;
typedef __attribute__((ext_vector_type(8))) float v8f;

// ---------------------------------------------------------------------------
// WMMA f32 GEMM: C[M,Nc] = A[M,K] @ B[K,Nc]  (all row-major, f32)
// Block = 128 threads (4 waves). Each wave computes a 16x64 output tile using
// V_WMMA_F32_16X16X4_F32, reusing the A fragment across 4 column tiles.
// A-row indices are clamped to Mreal-1 (padded rows never consumed).
// ---------------------------------------------------------------------------
__global__ __launch_bounds__(128) void gemm_wmma_f32(
    const float* __restrict__ A, const float* __restrict__ B,
    float* __restrict__ C, int Mreal, int K, int Nc) {
  const int lane = threadIdx.x & 31;
  const int wave = threadIdx.x >> 5;
  const int rowBase = (blockIdx.x * 4 + wave) * 16;
  const int colBase = blockIdx.y * 64;
  const int half = lane >> 4;   // K sub-pair selector (ISA 16x4 f32 layout)
  const int mIdx = lane & 15;

  int arow = rowBase + mIdx;
  if (arow >= Mreal) arow = Mreal - 1;
  const float* Aptr = A + (size_t)arow * K;

  v8f acc0 = {0.f,0.f,0.f,0.f,0.f,0.f,0.f,0.f};
  v8f acc1 = acc0, acc2 = acc0, acc3 = acc0;

  for (int k = 0; k < K; k += 4) {
    const int ka = k + 2 * half;
    v2f a; a.x = Aptr[ka]; a.y = Aptr[ka + 1];
    const float* B0 = B + (size_t)ka * Nc + colBase + mIdx;
    const float* B1 = B0 + Nc;
    v2f b0; b0.x = B0[0];  b0.y = B1[0];
    v2f b1; b1.x = B0[16]; b1.y = B1[16];
    v2f b2; b2.x = B0[32]; b2.y = B1[32];
    v2f b3; b3.x = B0[48]; b3.y = B1[48];
    acc0 = __builtin_amdgcn_wmma_f32_16x16x4_f32(false, a, false, b0, (short)0, acc0, false, false);
    acc1 = __builtin_amdgcn_wmma_f32_16x16x4_f32(false, a, false, b1, (short)0, acc1, false, false);
    acc2 = __builtin_amdgcn_wmma_f32_16x16x4_f32(false, a, false, b2, (short)0, acc2, false, false);
    acc3 = __builtin_amdgcn_wmma_f32_16x16x4_f32(false, a, false, b3, (short)0, acc3, false, false);
  }

  float* Cw = C + (size_t)rowBase * Nc + colBase;
  #pragma unroll
  for (int v = 0; v < 8; v++) {
    float* crow = Cw + (size_t)(v + 8 * half) * Nc + mIdx;
    crow[0]  = acc0[v];
    crow[16] = acc1[v];
    crow[32] = acc2[v];
    crow[48] = acc3[v];
  }
}

// ---------------------------------------------------------------------------
// Row-wise LayerNorm(+bias)(+residual) + ReLU over 128 columns. 1 wave / row.
// ---------------------------------------------------------------------------
__global__ __launch_bounds__(256) void ln_relu_kernel(
    const float* __restrict__ raw, const float* __restrict__ res,
    const float* __restrict__ bias, const float* __restrict__ w,
    const float* __restrict__ b, float* __restrict__ dst, int rows) {
  const int wave = threadIdx.x >> 5, lane = threadIdx.x & 31;
  const int row = blockIdx.x * 8 + wave;
  if (row >= rows) return;
  float4 v = ((const float4*)(raw + (size_t)row * 128))[lane];
  if (res) {
    float4 h = ((const float4*)(res + (size_t)row * 128))[lane];
    v.x += h.x; v.y += h.y; v.z += h.z; v.w += h.w;
  }
  if (bias) {
    float4 bb = ((const float4*)bias)[lane];
    v.x += bb.x; v.y += bb.y; v.z += bb.z; v.w += bb.w;
  }
  float s = v.x + v.y + v.z + v.w;
  for (int m = 1; m < 32; m <<= 1) s += __shfl_xor(s, m, 32);
  const float mean = s * (1.0f / 128.0f);
  const float dx = v.x - mean, dy = v.y - mean, dz = v.z - mean, dw = v.w - mean;
  float q = dx * dx + dy * dy + dz * dz + dw * dw;
  for (int m = 1; m < 32; m <<= 1) q += __shfl_xor(q, m, 32);
  const float inv = rsqrtf(q * (1.0f / 128.0f) + 1e-5f);
  const float4 wv = ((const float4*)w)[lane];
  const float4 bv = ((const float4*)b)[lane];
  float4 o;
  o.x = fmaxf(dx * inv * wv.x + bv.x, 0.f);
  o.y = fmaxf(dy * inv * wv.y + bv.y, 0.f);
  o.z = fmaxf(dz * inv * wv.z + bv.z, 0.f);
  o.w = fmaxf(dw * inv * wv.w + bv.w, 0.f);
  ((float4*)(dst + (size_t)row * 128))[lane] = o;
}

// ---------------------------------------------------------------------------
// Fold Wsrc/Wdst with attention vectors: vmat[j][0..3]=src heads, [4..7]=dst.
// ---------------------------------------------------------------------------
__global__ __launch_bounds__(1024) void prepv_kernel(
    const float* __restrict__ Wsrc, const float* __restrict__ Wdst,
    const float* __restrict__ as, const float* __restrict__ ad,
    float* __restrict__ vmat) {
  const int t = threadIdx.x;              // 0..1023
  const int which = t >> 9;               // 0 = src, 1 = dst
  const int idx = t & 511;
  const int j = idx >> 2, head = idx & 3;
  const float* W = which ? Wdst : Wsrc;
  const float* att = which ? ad : as;
  float s = 0.f;
  for (int k = 0; k < 128; k++) s += W[j * 512 + head * 128 + k] * att[head * 128 + k];
  vmat[j * 8 + which * 4 + head] = s;
}

// alpha[n][0..3] = h[n]·vsrc heads ; alpha[n][4..7] = h[n]·vdst heads
__global__ void alpha_kernel(const float* __restrict__ h,
                             const float* __restrict__ vmat,
                             float* __restrict__ alpha, int n) {
  const int t = blockIdx.x * 256 + threadIdx.x;
  if (t >= n * 8) return;
  const int node = t >> 3, f = t & 7;
  const float* hr = h + (size_t)node * 128;
  float s = 0.f;
  for (int j = 0; j < 128; j++) s += hr[j] * vmat[j * 8 + f];
  alpha[t] = s;
}

// ---------------------------------------------------------------------------
// CSR aggregation: one wave per destination node, 512-col accumulator in
// registers (4x float4 per lane, lane covers cols [lane*16, lane*16+16)).
// Per edge: per-edge head softmax (leaky 0.2) then weighted row accumulate.
// ---------------------------------------------------------------------------
__global__ __launch_bounds__(256) void agg_kernel(
    const float* __restrict__ Hsrc, const float* __restrict__ alpha,
    const int* __restrict__ row_ptr, const int* __restrict__ csr_src,
    float* __restrict__ agg, int n) {
  const int wave = threadIdx.x >> 5, lane = threadIdx.x & 31;
  const int node = blockIdx.x * 8 + wave;
  if (node >= n) return;
  const float ad0 = alpha[node * 8 + 4], ad1 = alpha[node * 8 + 5];
  const float ad2 = alpha[node * 8 + 6], ad3 = alpha[node * 8 + 7];
  const int head = lane >> 3;  // 16 cols/lane all inside one head
  float4 a0 = {0,0,0,0}, a1 = a0, a2 = a0, a3 = a0;
  const int beg = row_ptr[node], end = row_ptr[node + 1];
  for (int p = beg; p < end; p++) {
    const int s = csr_src[p];
    const float* al = alpha + (size_t)s * 8;
    float t0 = al[0] + ad0, t1 = al[1] + ad1, t2 = al[2] + ad2, t3 = al[3] + ad3;
    t0 = t0 > 0.f ? t0 : 0.2f * t0;  t1 = t1 > 0.f ? t1 : 0.2f * t1;
    t2 = t2 > 0.f ? t2 : 0.2f * t2;  t3 = t3 > 0.f ? t3 : 0.2f * t3;
    const float mx = fmaxf(fmaxf(t0, t1), fmaxf(t2, t3));
    const float e0 = __expf(t0 - mx), e1 = __expf(t1 - mx);
    const float e2 = __expf(t2 - mx), e3 = __expf(t3 - mx);
    const float inv = 1.f / (e0 + e1 + e2 + e3);
    const float wgt = ((head == 0) ? e0 : (head == 1) ? e1 : (head == 2) ? e2 : e3) * inv;
    const float4* sr = (const float4*)(Hsrc + (size_t)s * 512 + lane * 16);
    const float4 m0 = sr[0], m1 = sr[1], m2 = sr[2], m3 = sr[3];
    a0.x += wgt * m0.x; a0.y += wgt * m0.y; a0.z += wgt * m0.z; a0.w += wgt * m0.w;
    a1.x += wgt * m1.x; a1.y += wgt * m1.y; a1.z += wgt * m1.z; a1.w += wgt * m1.w;
    a2.x += wgt * m2.x; a2.y += wgt * m2.y; a2.z += wgt * m2.z; a2.w += wgt * m2.w;
    a3.x += wgt * m3.x; a3.y += wgt * m3.y; a3.z += wgt * m3.z; a3.w += wgt * m3.w;
  }
  float4* dst = (float4*)(agg + (size_t)node * 512 + lane * 16);
  dst[0] = a0; dst[1] = a1; dst[2] = a2; dst[3] = a3;
}

// --------------------------- CSR construction -----------------------------
__global__ void zero_i32(int* p, int n) {
  int t = blockIdx.x * 256 + threadIdx.x; if (t < n) p[t] = 0;
}
__global__ void copy_i32(int* d, const int* s, int n) {
  int t = blockIdx.x * 256 + threadIdx.x; if (t < n) d[t] = s[t];
}
__global__ void count_kernel(const long long* __restrict__ ei, int* deg, int E, int n) {
  int e = blockIdx.x * 256 + threadIdx.x;
  if (e >= E + n) return;
  int dst = (e < E) ? (int)ei[E + e] : (e - E);
  atomicAdd(&deg[dst], 1);
}
__global__ __launch_bounds__(1024) void scan_kernel(const int* __restrict__ deg,
                                                    int* __restrict__ row_ptr, int n) {
  __shared__ int sm[1024];
  __shared__ int carry;
  const int tid = threadIdx.x;
  if (tid == 0) carry = 0;
  __syncthreads();
  for (int base = 0; base < n; base += 1024) {
    const int i = base + tid;
    const int v = (i < n) ? deg[i] : 0;
    sm[tid] = v;
    __syncthreads();
    for (int off = 1; off < 1024; off <<= 1) {
      int t = (tid >= off) ? sm[tid - off] : 0;
      __syncthreads();
      sm[tid] += t;
      __syncthreads();
    }
    const int incl = sm[tid];
    const int c = carry;
    if (i < n) row_ptr[i] = c + incl - v;
    __syncthreads();
    if (tid == 1023) carry = c + sm[1023];
    __syncthreads();
  }
  if (tid == 0) row_ptr[n] = carry;
}
__global__ void scatter_kernel(const long long* __restrict__ ei, int* cursor,
                               int* __restrict__ csr_src, int E, int n) {
  int e = blockIdx.x * 256 + threadIdx.x;
  if (e >= E + n) return;
  int src, dst;
  if (e < E) { src = (int)ei[e]; dst = (int)ei[E + e]; }
  else       { src = dst = e - E; }
  const int pos = atomicAdd(&cursor[dst], 1);
  csr_src[pos] = src;
}

// ------------------------------ pooling / MLP ------------------------------
__global__ void pool_kernel(const float* __restrict__ h, const long long* __restrict__ batch,
                            float* __restrict__ g, int n) {
  int t = blockIdx.x * 256 + threadIdx.x;
  if (t >= n * 128) return;
  const int node = t >> 7, c = t & 127;
  const int bg = (int)batch[node];
  // h >= 0 after ReLU => uint bit pattern order == float order
  atomicMax((unsigned int*)&g[bg * 128 + c], __float_as_uint(h[(size_t)node * 128 + c]));
}
__global__ void mlp_kernel(const float* __restrict__ X, const float* __restrict__ W,
                           const float* __restrict__ bias, float* __restrict__ Y,
                           int M, int K, int Nc, int doRelu) {
  int t = blockIdx.x * 256 + threadIdx.x;
  if (t >= M * Nc) return;
  const int m = t / Nc, oc = t % Nc;
  const float* x = X + (size_t)m * K;
  float s = bias[oc];
  for (int k = 0; k < K; k++) s += x[k] * W[(size_t)k * Nc + oc];
  if (doRelu) s = fmaxf(s, 0.f);
  Y[t] = s;
}

// ---------------------------------------------------------------------------
static inline void launch_gemm(const float* A, const float* B, float* C,
                               int Mreal, int K, int Nc, hipStream_t stream) {
  const int tilesM = (Mreal + 15) / 16;
  dim3 grid((tilesM + 3) / 4, Nc / 64);
  gemm_wmma_f32<<<grid, 128, 0, stream>>>(A, B, C, Mreal, K, Nc);
}

extern "C" void kernel_launch(void* const* d_in, const int* in_sizes, int n_in,
                              void* d_out, int out_size, void* d_ws, size_t ws_size,
                              hipStream_t stream) {
  const float*     x        = (const float*)d_in[0];
  const long long* edge_idx = (const long long*)d_in[1];
  const long long* batch    = (const long long*)d_in[2];
  const float* W_in   = (const float*)d_in[3];
  const float* b_in   = (const float*)d_in[4];
  const float* ln_in_w= (const float*)d_in[5];
  const float* ln_in_b= (const float*)d_in[6];
  const float* Wsrc   = (const float*)d_in[7];
  const float* Wdst   = (const float*)d_in[8];
  const float* att_src= (const float*)d_in[9];
  const float* att_dst= (const float*)d_in[10];
  const float* Wout   = (const float*)d_in[11];
  const float* bout   = (const float*)d_in[12];
  const float* ln_w   = (const float*)d_in[13];
  const float* ln_b   = (const float*)d_in[14];
  const float* Wp1    = (const float*)d_in[15];
  const float* bp1    = (const float*)d_in[16];
  const float* Wp2    = (const float*)d_in[17];
  const float* bp2    = (const float*)d_in[18];
  float* out = (float*)d_out;

  const int N = in_sizes[2];          // 10000
  const int E = in_sizes[1] / 2;      // 160000
  const int E2 = E + N;
  const int D_IN = 256, H = 128, HD = 512, L = 16, OUTD = 1024, G = 8;
  const int tilesM = (N + 15) / 16;
  const int NPAD = ((tilesM + 3) / 4) * 64;   // rows writable by GEMM tiling

  // carve workspace (256B aligned)
  char* wp = (char*)d_ws;
  auto alloc = [&](size_t bytes) -> char* {
    char* p = wp; wp += (bytes + 255) & ~(size_t)255; return p;
  };
  float* hA    = (float*)alloc((size_t)NPAD * H * 4);
  float* hB    = (float*)alloc((size_t)NPAD * H * 4);
  float* raw   = (float*)alloc((size_t)NPAD * H * 4);
  float* Hs    = (float*)alloc((size_t)NPAD * HD * 4);
  float* agg   = (float*)alloc((size_t)NPAD * HD * 4);
  float* alpha = (float*)alloc((size_t)NPAD * 8 * 4);
  float* vmat  = (float*)alloc(128 * 8 * 4);
  int*   deg   = (int*)alloc((size_t)N * 4);
  int*   rowp  = (int*)alloc((size_t)(N + 1) * 4);
  int*   curs  = (int*)alloc((size_t)N * 4);
  int*   csr   = (int*)alloc((size_t)E2 * 4);
  float* g     = (float*)alloc((size_t)G * H * 4);
  float* g2    = (float*)alloc((size_t)G * OUTD * 4);

  // CSR build (e_dst fixed across layers)
  zero_i32<<<(N + 255) / 256, 256, 0, stream>>>(deg, N);
  count_kernel<<<(E2 + 255) / 256, 256, 0, stream>>>(edge_idx, deg, E, N);
  scan_kernel<<<1, 1024, 0, stream>>>(deg, rowp, N);
  copy_i32<<<(N + 255) / 256, 256, 0, stream>>>(curs, rowp, N);
  scatter_kernel<<<(E2 + 255) / 256, 256, 0, stream>>>(edge_idx, curs, csr, E, N);

  // input projection + LN + ReLU
  launch_gemm(x, W_in, raw, N, D_IN, H, stream);
  ln_relu_kernel<<<(N + 7) / 8, 256, 0, stream>>>(raw, nullptr, b_in, ln_in_w, ln_in_b, hA, N);

  float* h = hA; float* hn = hB;
  for (int l = 0; l < L; l++) {
    const float* Ws = Wsrc + (size_t)l * H * HD;
    const float* Wd = Wdst + (size_t)l * H * HD;
    prepv_kernel<<<1, 1024, 0, stream>>>(Ws, Wd, att_src + (size_t)l * 512,
                                         att_dst + (size_t)l * 512, vmat);
    launch_gemm(h, Ws, Hs, N, H, HD, stream);
    alpha_kernel<<<(N * 8 + 255) / 256, 256, 0, stream>>>(h, vmat, alpha, N);
    agg_kernel<<<(N + 7) / 8, 256, 0, stream>>>(Hs, alpha, rowp, csr, agg, N);
    launch_gemm(agg, Wout + (size_t)l * HD * H, raw, N, HD, H, stream);
    ln_relu_kernel<<<(N + 7) / 8, 256, 0, stream>>>(raw, h, bout + (size_t)l * H,
                                                    ln_w + (size_t)l * H,
                                                    ln_b + (size_t)l * H, hn, N);
    float* t = h; h = hn; hn = t;
  }

  // global max pool (+0.0f identity valid since h >= 0 post-ReLU)
  zero_i32<<<(G * H + 255) / 256, 256, 0, stream>>>((int*)g, G * H);
  pool_kernel<<<(N * 128 + 255) / 256, 256, 0, stream>>>(h, batch, g, N);
  mlp_kernel<<<(G * OUTD + 255) / 256, 256, 0, stream>>>(g, Wp1, bp1, g2, G, H, OUTD, 1);
  mlp_kernel<<<(G * OUTD + 255) / 256, 256, 0, stream>>>(g2, Wp2, bp2, out, G, OUTD, OUTD, 0);
}